// WaveletLowpass_20538533609640
// MI455X (gfx1250) — compile-verified
//
#include <hip/hip_runtime.h>
#include <stdint.h>

// Haar LLL 2x2x2 downsample, fp32.
// in : (8,160,160,160) f32     out: (8,80,80,80) f32
// scale = 0.5^1.5
#define HAAR_SCALE 0.35355339059327373f

typedef float v2f __attribute__((ext_vector_type(2)));
typedef float v4f __attribute__((ext_vector_type(4)));

// Block geometry:
//   320 threads (10 wave32).
//   blockIdx.x -> (b, dz, g) with g = group of 8 output rows (dy in [8g, 8g+8)).
//   Input tile: d in {2dz, 2dz+1}, h in [16g, 16g+16), w in [0,160)
//             = 2 contiguous chunks of 16*160 floats (10 KB each) -> 20 KB LDS.
//   Async DMA: each lane issues 4x global_load_async_to_lds_b128 (16 B per op).
//   Compute : each thread produces 2 adjacent outputs along W from 4x ds_load_b128.
__global__ __launch_bounds__(320)
void haar_lll_kernel(const float* __restrict__ x, float* __restrict__ out)
{
    __shared__ __attribute__((aligned(16))) float tile[2 * 16 * 160]; // 20 KB

    const int t   = (int)threadIdx.x;      // 0..319
    const int bid = (int)blockIdx.x;       // 0..6399
    const int g   = bid % 10;              // row-group
    const int dz  = (bid / 10) % 80;       // output depth
    const int b   = bid / 800;             // batch

    const int d0 = dz * 2;
    const int h0 = g * 16;

    // ---------------- async global -> LDS staging ----------------
    const uint32_t lds_base = (uint32_t)(uintptr_t)&tile[0];
#pragma unroll
    for (int dd = 0; dd < 2; ++dd) {
        // contiguous 16x160 f32 chunk for this input depth slice
        const float* chunk =
            x + ((size_t)((b * 160 + d0 + dd) * 160 + h0) * 160);
#pragma unroll
        for (int s = 0; s < 2; ++s) {
            const uint32_t off  = (uint32_t)(t + s * 320) * 16u;       // byte offset in chunk
            const uint32_t ldsa = lds_base + (uint32_t)dd * 10240u + off;
            // CDNA5 async copy: copies 16 bytes from global chunk at byte offset off to the LDS address in ldsa, ASYNCcnt-tracked
            asm volatile("global_load_async_to_lds_b128 %0, %1, %2"
                         :
                         : "v"(ldsa), "v"(off), "s"(chunk)
                         : "memory");
        }
    }
    // wait for this wave's async copies, then make LDS visible workgroup-wide
    asm volatile("s_wait_asynccnt 0" ::: "memory");
    __syncthreads();

    // ---------------- 2x2x2 reduction from LDS ----------------
    const int q  = t % 40;   // output w pair: wo = 2q, 2q+1  (input w = 4q..4q+3)
    const int dy = t / 40;   // 0..7 within row group

    const int r0 = (2 * dy) * 160 + 4 * q;          // h row 2dy,   depth d0
    const int r1 = (2 * dy + 1) * 160 + 4 * q;      // h row 2dy+1, depth d0

    const v4f a = *(const v4f*)&tile[r0];
    const v4f c = *(const v4f*)&tile[r1];
    const v4f e = *(const v4f*)&tile[2560 + r0];    // depth d0+1
    const v4f f = *(const v4f*)&tile[2560 + r1];

    const v4f s4 = (a + c) + (e + f);               // sum over d,h
    const float o0 = (s4.x + s4.y) * HAAR_SCALE;    // sum over w pair 0
    const float o1 = (s4.z + s4.w) * HAAR_SCALE;    // sum over w pair 1

    // output index: (b, dz, 8g+dy, 2q) ; write 8 B, non-temporal (never re-read)
    const size_t oidx =
        (((size_t)b * 80 + dz) * 80 + (size_t)(h0 / 2 + dy)) * 80 + (size_t)(2 * q);
    v2f ov; ov.x = o0; ov.y = o1;
    __builtin_nontemporal_store(ov, (v2f*)(out + oidx));
}

extern "C" void kernel_launch(void* const* d_in, const int* in_sizes, int n_in,
                              void* d_out, int out_size, void* d_ws, size_t ws_size,
                              hipStream_t stream)
{
    (void)in_sizes; (void)n_in; (void)out_size; (void)d_ws; (void)ws_size;
    const float* x  = (const float*)d_in[0];
    float*       out = (float*)d_out;

    // 8 batches * 80 output depths * 10 row-groups = 6400 blocks, 320 threads each.
    dim3 grid(6400), block(320);
    hipLaunchKernelGGL(haar_lll_kernel, grid, block, 0, stream, x, out);
}